// GraphSAGE_90658169683982
// MI455X (gfx1250) — compile-verified
//
#include <hip/hip_runtime.h>

// GraphSAGE forward for MI455X (gfx1250), fp32 WMMA path.
// d_in order: x, edge_src, edge_dst, W_self1, W_neigh1, b1,
//             W_self2, W_neigh2, b2, W_out, b_out

typedef float v2f __attribute__((ext_vector_type(2)));
typedef float v8f __attribute__((ext_vector_type(8)));

#define FEAT 128
#define MT 5   // row tiles per wave; 50000 = (16*MT) * 625

__global__ __launch_bounds__(256) void sage_zero_f32(float* __restrict__ p, long n) {
  long i = (long)blockIdx.x * blockDim.x + threadIdx.x;
  if (i < n) p[i] = 0.0f;
}

// One wave per edge: 32 lanes x float4 = 128 features. Gather h[src], atomic
// scatter-add into agg[dst]. Lane 0 optionally accumulates in-degree.
__global__ __launch_bounds__(256) void sage_edge_scatter(
    const float* __restrict__ h, const int* __restrict__ src,
    const int* __restrict__ dst, float* __restrict__ agg,
    float* __restrict__ deg, int E, int accum_deg)
{
  int e = (int)((blockIdx.x * 256 + threadIdx.x) >> 5);
  if (e >= E) return;
  int lane = threadIdx.x & 31;
  // edge index is wave-uniform -> scalarize the src/dst loads
  int s = __builtin_amdgcn_readfirstlane(src[e]);
  int d = __builtin_amdgcn_readfirstlane(dst[e]);

  const float4 v = *reinterpret_cast<const float4*>(h + (long)s * FEAT + lane * 4);
  float* b = agg + (long)d * FEAT + lane * 4;
  atomicAdd(b + 0, v.x);
  atomicAdd(b + 1, v.y);
  atomicAdd(b + 2, v.z);
  atomicAdd(b + 3, v.w);
  if (accum_deg && lane == 0) atomicAdd(deg + d, 1.0f);
}

// Fused SAGE layer GEMM using V_WMMA_F32_16X16X4_F32 (wave32).
// Out[M,NOUT] = act( H[M,K]·Wself[K,NOUT] + (AGG[M,K]*deg^-1)·Wneigh[K,NOUT] + bias )
// Grid.x = M/(16*MT); one wave per 16-wide column tile (blockDim = 32*NOUT/16);
// each wave computes MT stacked 16x16 row tiles, reusing each B fragment MT times.
// K/NOUT are compile-time so tile strides become load-immediate offsets.
// Requires M%(16*MT)==0, K%4==0, NOUT%16==0.
template <bool HAS_AGG, bool RELU, int K, int NOUT>
__global__ __launch_bounds__(256) void sage_gemm_wmma(
    const float* __restrict__ H, const float* __restrict__ AGG,
    const float* __restrict__ deg,
    const float* __restrict__ Wself, const float* __restrict__ Wneigh,
    const float* __restrict__ bias, float* __restrict__ Out)
{
  const int lane  = threadIdx.x & 31;
  const int wv    = threadIdx.x >> 5;
  const int half  = lane >> 4;   // 0: K pair {0,1}; 1: K pair {2,3}
  const int l     = lane & 15;
  const int mbase = blockIdx.x * (16 * MT);
  const int nbase = wv * 16;

  // C/D layout: element (vgpr i, lane) = C[tile_m + i + 8*half][nbase + l].
  v8f acc[MT];
  const float bval = bias[nbase + l];
#pragma unroll
  for (int t = 0; t < MT; ++t)
#pragma unroll
    for (int i = 0; i < 8; ++i) acc[t][i] = bval;

  const float* hrow = H + (long)(mbase + l) * K;  // A row for this lane, tile 0
  const float* arow = nullptr;
  float scale[MT];
  if (HAS_AGG) {
    arow = AGG + (long)(mbase + l) * K;
#pragma unroll
    for (int t = 0; t < MT; ++t)                   // fold mean-normalization
      scale[t] = 1.0f / fmaxf(deg[mbase + t * 16 + l], 1.0f);
  }

  for (int k0 = 0; k0 < K; k0 += 4) {
    const int ka = k0 + 2 * half;
    // B 4x16 fragment: lane l = column, VGPR0/1 = rows ka, ka+1
    v2f b;
    b.x = Wself[(long)ka * NOUT + nbase + l];
    b.y = Wself[(long)(ka + 1) * NOUT + nbase + l];
#pragma unroll
    for (int t = 0; t < MT; ++t) {
      // A 16x4 fragment: lane l = row; tile stride 16*K floats is an imm offset
      v2f a = *reinterpret_cast<const v2f*>(hrow + t * (16 * K) + ka);
      acc[t] = __builtin_amdgcn_wmma_f32_16x16x4_f32(false, a, false, b,
                                                     (short)0, acc[t], false, false);
    }
    if (HAS_AGG) {
      v2f b2;
      b2.x = Wneigh[(long)ka * NOUT + nbase + l];
      b2.y = Wneigh[(long)(ka + 1) * NOUT + nbase + l];
#pragma unroll
      for (int t = 0; t < MT; ++t) {
        v2f a2 = *reinterpret_cast<const v2f*>(arow + t * (16 * K) + ka);
        a2.x *= scale[t];
        a2.y *= scale[t];
        acc[t] = __builtin_amdgcn_wmma_f32_16x16x4_f32(false, a2, false, b2,
                                                       (short)0, acc[t], false, false);
      }
    }
  }

#pragma unroll
  for (int t = 0; t < MT; ++t) {
#pragma unroll
    for (int i = 0; i < 8; ++i) {
      float v = acc[t][i];
      if (RELU) v = fmaxf(v, 0.0f);
      Out[(long)(mbase + t * 16 + i + 8 * half) * NOUT + nbase + l] = v;
    }
  }
}

extern "C" void kernel_launch(void* const* d_in, const int* in_sizes, int n_in,
                              void* d_out, int out_size, void* d_ws, size_t ws_size,
                              hipStream_t stream) {
  const float* x    = (const float*)d_in[0];
  const int*   esrc = (const int*)d_in[1];
  const int*   edst = (const int*)d_in[2];
  const float* Ws1  = (const float*)d_in[3];
  const float* Wn1  = (const float*)d_in[4];
  const float* b1   = (const float*)d_in[5];
  const float* Ws2  = (const float*)d_in[6];
  const float* Wn2  = (const float*)d_in[7];
  const float* b2   = (const float*)d_in[8];
  const float* Wo   = (const float*)d_in[9];
  const float* bo   = (const float*)d_in[10];
  float* out = (float*)d_out;

  const int N = in_sizes[0] / FEAT;  // 50000 (divisible by 16*MT = 80)
  const int E = in_sizes[1];         // 800000

  // Workspace layout (fp32): agg[N*128] | deg[N] | h1[N*128] | h2[N*128]
  float* agg = (float*)d_ws;
  float* deg = agg + (long)N * FEAT;
  float* h1  = deg + N;
  float* h2  = h1 + (long)N * FEAT;

  const long nAgg  = (long)N * FEAT;
  const int  sgrid = (E + 7) / 8;       // 8 waves (edges) per 256-thread block
  const int  mgrid = N / (16 * MT);     // 625

  // Layer 1
  sage_zero_f32<<<(int)((nAgg + N + 255) / 256), 256, 0, stream>>>(agg, nAgg + N);
  sage_edge_scatter<<<sgrid, 256, 0, stream>>>(x, esrc, edst, agg, deg, E, 1);
  sage_gemm_wmma<true, true, FEAT, FEAT>
      <<<mgrid, 256, 0, stream>>>(x, agg, deg, Ws1, Wn1, b1, h1);
  // Layer 2 (deg unchanged: depends only on edge_dst)
  sage_zero_f32<<<(int)((nAgg + 255) / 256), 256, 0, stream>>>(agg, nAgg);
  sage_edge_scatter<<<sgrid, 256, 0, stream>>>(h1, esrc, edst, agg, deg, E, 0);
  sage_gemm_wmma<true, true, FEAT, FEAT>
      <<<mgrid, 256, 0, stream>>>(h1, agg, deg, Ws2, Wn2, b2, h2);
  // Output projection: 64 cols -> 4 waves per row tile group, no agg, no relu
  sage_gemm_wmma<false, false, FEAT, 64>
      <<<mgrid, 128, 0, stream>>>(h2, nullptr, nullptr, Wo, nullptr, bo, out);
}